// Histogram_Matching_51745765982911
// MI455X (gfx1250) — compile-verified
//
#include <hip/hip_runtime.h>
#include <hip/hip_bf16.h>

// ---------------------------------------------------------------------------
// Histogram matching (8,3,1024,1024) f32.
//   Phase 1: 48 per-channel 256-bin histograms (dst + ref), LDS-privatized.
//   Phase 2: CDF via WMMA f32 16x16x4: cdf = H @ triu(ones)  (faithful to the
//            reference's "triu bmm == cumsum").
//   Phase 3: transfer table per channel (first-match scan), fold /255 into LUT.
//   Phase 4: apply LUT (bug-faithful b*c row indexing), float4 streaming.
// ---------------------------------------------------------------------------

#define NCHAN 24          // B*C = 8*3 dst channels
#define NHIST 48          // dst + ref histograms
#define BINS 256
#define PIX_PER_CHAN (1024 * 1024)
#define HIST_BLOCKS_PER_CHAN 32
#define PIX_PER_HBLOCK (PIX_PER_CHAN / HIST_BLOCKS_PER_CHAN)   // 32768
#define APPLY_BLOCKS_PER_CHAN 32
#define PIX_PER_ABLOCK (PIX_PER_CHAN / APPLY_BLOCKS_PER_CHAN)  // 32768

typedef float v2f __attribute__((ext_vector_type(2)));
typedef float v8f __attribute__((ext_vector_type(8)));

__device__ __forceinline__ int bin_of(float x01) {
    // d = x*255; idx = trunc(d * 256/255); clip [0,255]   (matches jnp)
    float d = x01 * 255.0f;
    int idx = (int)(d * (256.0f / 255.0f));
    idx = idx < 0 ? 0 : (idx > 255 ? 255 : idx);
    return idx;
}

// ---------------- Phase 0: zero the histogram accumulators ----------------
__global__ __launch_bounds__(256) void zero_hist_kernel(unsigned int* __restrict__ hist) {
    hist[blockIdx.x * 256 + threadIdx.x] = 0u;
}

// ---------------- Phase 1: histograms (48 channels) -----------------------
__global__ __launch_bounds__(256) void hist_kernel(const float* __restrict__ dst,
                                                   const float* __restrict__ ref,
                                                   unsigned int* __restrict__ hist) {
    __shared__ unsigned int lh[8 * BINS];   // one sub-histogram per wave32
    const int tid  = threadIdx.x;
    const int wave = tid >> 5;

    #pragma unroll
    for (int i = tid; i < 8 * BINS; i += 256) lh[i] = 0u;
    __syncthreads();

    const int chan = blockIdx.y;            // 0..47
    const float* src = (chan < NCHAN)
        ? dst + (size_t)chan * PIX_PER_CHAN
        : ref + (size_t)(chan - NCHAN) * PIX_PER_CHAN;

    const float4* s4 = (const float4*)(src + (size_t)blockIdx.x * PIX_PER_HBLOCK);
    const int n4 = PIX_PER_HBLOCK / 4;
    unsigned int* mywave = &lh[wave * BINS];

    for (int i = tid; i < n4; i += 256) {
        float4 v = s4[i];
        atomicAdd(&mywave[bin_of(v.x)], 1u);
        atomicAdd(&mywave[bin_of(v.y)], 1u);
        atomicAdd(&mywave[bin_of(v.z)], 1u);
        atomicAdd(&mywave[bin_of(v.w)], 1u);
    }
    __syncthreads();

    // tid == bin (blockDim.x == 256)
    unsigned int s = 0;
    #pragma unroll
    for (int w = 0; w < 8; ++w) s += lh[w * BINS + tid];
    atomicAdd(&hist[chan * BINS + tid], s);
}

// ---------------- Phase 2: CDF via WMMA (cdf = H @ triu(ones)) ------------
// One wave per 16x16 output tile. Grid = (16 N-tiles, 3 M-tiles).
// A (16x4 f32): lanes 0-15 -> M=lane, VGPR0/1 = K0/K1; lanes 16-31 -> K2/K3.
// B (4x16 f32): VGPR0 = row K0 (lanes 0-15) / K2 (lanes 16-31); VGPR1 = K1/K3.
// C/D (16x16 f32): VGPR i: lanes 0-15 -> M=i, lanes 16-31 -> M=i+8.
__global__ __launch_bounds__(32) void cdf_wmma_kernel(const unsigned int* __restrict__ hist,
                                                      float* __restrict__ cdf) {
    const int nt   = blockIdx.x;        // 0..15  (N tile)
    const int mt   = blockIdx.y;        // 0..2   (M tile)
    const int lane = threadIdx.x;       // 0..31
    const int half = lane >> 4;         // 0 or 1
    const int l15  = lane & 15;

    const int   row   = mt * 16 + l15;            // channel (A row), 0..47
    const int   ncol  = nt * 16 + l15;            // global output column
    const float scale = 1.0f / 1048576.0f;        // exact L1 norm (pixel count)
    const unsigned int* hrow = hist + row * BINS;

    v8f c = {0.0f, 0.0f, 0.0f, 0.0f, 0.0f, 0.0f, 0.0f, 0.0f};
    const int kmax = nt * 16 + 16;                // triu: K <= tile's top column

    for (int k = 0; k < kmax; k += 4) {
        const int ka = k + half * 2;
        v2f a, b;
        a.x = (float)hrow[ka]     * scale;
        a.y = (float)hrow[ka + 1] * scale;
        b.x = (ka     <= ncol) ? 1.0f : 0.0f;     // triu(ones) block
        b.y = (ka + 1 <= ncol) ? 1.0f : 0.0f;
        c = __builtin_amdgcn_wmma_f32_16x16x4_f32(
                /*neg_a=*/false, a, /*neg_b=*/false, b,
                /*c_mod=*/(short)0, c, /*reuse_a=*/false, /*reuse_b=*/false);
    }

    #pragma unroll
    for (int i = 0; i < 8; ++i) {
        const int r = mt * 16 + i + half * 8;
        cdf[r * BINS + ncol] = c[i];
    }
}

// ---------------- Phase 3: transfer table -> LUT (pre-divided by 255) -----
__global__ __launch_bounds__(256) void table_kernel(const float* __restrict__ cdf,
                                                    float* __restrict__ lut) {
    __shared__ float hd[BINS];
    __shared__ float hr[BINS];
    const int m = blockIdx.x;           // dst channel 0..23
    const int i = threadIdx.x;
    hd[i] = cdf[m * BINS + i];
    hr[i] = cdf[(NCHAN + m) * BINS + i];
    __syncthreads();

    int t;
    if (i == 0) {
        t = 0;
    } else if (i == BINS - 1) {
        t = BINS - 1;
    } else {
        t = i;
        const float v = hd[i];
        for (int j = 1; j < BINS; ++j) {
            if (hr[j - 1] <= v && v <= hr[j]) { t = j; break; }   // first match
        }
    }
    lut[m * BINS + i] = (float)t / 255.0f;
}

// ---------------- Phase 4: apply LUT --------------------------------------
__global__ __launch_bounds__(256) void apply_kernel(const float* __restrict__ dst,
                                                    const float* __restrict__ lut,
                                                    float* __restrict__ out) {
    __shared__ float l[BINS];
    const int ch  = blockIdx.y;                       // output channel 0..23
    const int row = (ch / 3) * (ch % 3);              // bug-faithful b*c indexing
    l[threadIdx.x] = lut[row * BINS + threadIdx.x];
    __syncthreads();

    const size_t base = (size_t)ch * PIX_PER_CHAN + (size_t)blockIdx.x * PIX_PER_ABLOCK;
    const float4* s4 = (const float4*)(dst + base);
    float4*       o4 = (float4*)(out + base);
    const int n4 = PIX_PER_ABLOCK / 4;

    for (int i = threadIdx.x; i < n4; i += 256) {
        float4 v = s4[i];
        float4 r;
        int p0 = (int)(v.x * 255.0f); p0 = p0 < 0 ? 0 : (p0 > 255 ? 255 : p0);
        int p1 = (int)(v.y * 255.0f); p1 = p1 < 0 ? 0 : (p1 > 255 ? 255 : p1);
        int p2 = (int)(v.z * 255.0f); p2 = p2 < 0 ? 0 : (p2 > 255 ? 255 : p2);
        int p3 = (int)(v.w * 255.0f); p3 = p3 < 0 ? 0 : (p3 > 255 ? 255 : p3);
        r.x = l[p0]; r.y = l[p1]; r.z = l[p2]; r.w = l[p3];
        o4[i] = r;
    }
}

// ---------------------------------------------------------------------------
extern "C" void kernel_launch(void* const* d_in, const int* in_sizes, int n_in,
                              void* d_out, int out_size, void* d_ws, size_t ws_size,
                              hipStream_t stream) {
    (void)in_sizes; (void)n_in; (void)out_size; (void)ws_size;
    const float* dst = (const float*)d_in[0];
    const float* ref = (const float*)d_in[1];
    float* out = (float*)d_out;

    unsigned int* hist = (unsigned int*)d_ws;                                  // 48*256 u32
    float* cdf = (float*)((char*)d_ws + (size_t)NHIST * BINS * 4);             // 48*256 f32
    float* lut = (float*)((char*)d_ws + (size_t)NHIST * BINS * 8);             // 24*256 f32

    zero_hist_kernel<<<dim3(NHIST), 256, 0, stream>>>(hist);
    hist_kernel<<<dim3(HIST_BLOCKS_PER_CHAN, NHIST), 256, 0, stream>>>(dst, ref, hist);
    cdf_wmma_kernel<<<dim3(16, 3), 32, 0, stream>>>(hist, cdf);
    table_kernel<<<dim3(NCHAN), 256, 0, stream>>>(cdf, lut);
    apply_kernel<<<dim3(APPLY_BLOCKS_PER_CHAN, NCHAN), 256, 0, stream>>>(dst, lut, out);
}